// QuantizedEmbedding_3418793967992
// MI455X (gfx1250) — compile-verified
//
#include <hip/hip_runtime.h>
#include <cstdint>

// QuantizedEmbedding: gather + per-row asymmetric int8 dequant.
//   out[gr, d] = (q[idx[gr], d] - zp[idx[gr]]) * scale[idx[gr]]
// Pure bandwidth problem (~168 MB moved, ~67 MFLOP -> no matrix content, WMMA
// inapplicable). Optimized data-movement path for MI455X: async global->LDS
// staging (ASYNCcnt pipeline), double buffered, scalar (SMEM) metadata loads,
// fully coalesced b128 stores, fma-based dequant.

#define DIM            1024
#define ROWS_PER_STAGE 4                         // 4 rows * 1 KB = 4 KB / stage
#define STAGES         4                         // stages per block
#define ROWS_PER_BLOCK (ROWS_PER_STAGE * STAGES) // 16
#define THREADS        256                       // 8 wave32s

__global__ __launch_bounds__(THREADS)
void dequant_gather_async(const int* __restrict__ idx,
                          const int8_t* __restrict__ qw,
                          const float* __restrict__ scales,
                          const float* __restrict__ zps,
                          float* __restrict__ out,
                          int n_rows)
{
    // double-buffered staging area: 2 x 4 rows x 1024 B = 8 KB
    __shared__ uint8_t stage[2][ROWS_PER_STAGE * DIM];

    const int t = threadIdx.x;
    // Row slot (t>>6) is constant within each wave32; make that explicit so
    // the compiler uses scalar branches and s_load for idx/scales/zps.
    const int s    = __builtin_amdgcn_readfirstlane(t >> 6);  // 0..3
    const int u    = t & 63;                                  // lane-in-rowgroup
    const int row0 = __builtin_amdgcn_readfirstlane(blockIdx.x) * ROWS_PER_BLOCK;

    // Issue the async copy of stage j: each lane stages 16 contiguous bytes of
    // its row into LDS (256 lanes cover the whole 4 KB stage buffer).
    auto issue = [&](int j) {
        const int gr = row0 + j * ROWS_PER_STAGE + s;   // scalar
        if (gr < n_rows) {                              // scalar branch
            const int token = idx[gr];                  // s_load_b32
            const uint32_t voff = (uint32_t)token * (uint32_t)DIM + (uint32_t)(u << 4);
            const uint32_t loff =
                (uint32_t)(uint64_t)(const void*)&stage[j & 1][t << 4];
            asm volatile("global_load_async_to_lds_b128 %0, %1, %2"
                         :: "v"(loff), "v"(voff), "s"(qw)
                         : "memory");
        }
    };

    issue(0);

    for (int j = 0; j < STAGES; ++j) {
        const int gr = row0 + j * ROWS_PER_STAGE + s;   // scalar

        // Keep the pipeline full: kick off stage j+1 before consuming stage j.
        const bool next_issued =
            (j + 1 < STAGES) && ((row0 + (j + 1) * ROWS_PER_STAGE + s) < n_rows);
        if (next_issued) {
            issue(j + 1);
            // async ops complete in order: <=1 outstanding => stage j landed
            asm volatile("s_wait_asynccnt %0" :: "i"(1) : "memory");
        } else {
            asm volatile("s_wait_asynccnt %0" :: "i"(0) : "memory");
        }
        __syncthreads();  // every wave waited its own asynccnt -> buffer valid

        if (gr < n_rows) {
            const int   token = idx[gr];                // s_load_b32
            const float sc    = scales[token];          // s_load_b32
            const float zp    = zps[token];             // s_load_b32
            const float bias  = -zp * sc;               // (q - zp)*sc == q*sc + bias
            const int*  lbuf  = (const int*)&stage[j & 1][s * DIM];
            float*      orow  = out + (size_t)gr * DIM;

            #pragma unroll
            for (int k = 0; k < 4; ++k) {
                // lanes read consecutive dwords -> bank-conflict-free ds_load_b32
                const int w = lbuf[k * 64 + u];
                float4 f;
                f.x = __builtin_fmaf((float)(int8_t)( w        ), sc, bias);
                f.y = __builtin_fmaf((float)(int8_t)( w >>  8  ), sc, bias);
                f.z = __builtin_fmaf((float)(int8_t)( w >> 16  ), sc, bias);
                f.w = __builtin_fmaf((float)(int8_t)( w >> 24  ), sc, bias);
                // 32 lanes x 16 B contiguous -> fully coalesced b128 store
                *(float4*)(orow + k * 256 + u * 4) = f;
            }
        }
        __syncthreads();  // buffer (j&1) free before issue(j+2) reuses it
    }
}

extern "C" void kernel_launch(void* const* d_in, const int* in_sizes, int n_in,
                              void* d_out, int out_size, void* d_ws, size_t ws_size,
                              hipStream_t stream) {
    const int*    idx = (const int*)d_in[0];
    const int8_t* qw  = (const int8_t*)d_in[1];
    const float*  sc  = (const float*)d_in[2];
    const float*  zp  = (const float*)d_in[3];
    float*        out = (float*)d_out;

    const int n_rows = in_sizes[0];  // B*S = 32768 token indices
    const int blocks = (n_rows + ROWS_PER_BLOCK - 1) / ROWS_PER_BLOCK;

    dequant_gather_async<<<blocks, THREADS, 0, stream>>>(idx, qw, sc, zp, out, n_rows);
}